// CFSM_56762287784215
// MI455X (gfx1250) — compile-verified
//
#include <hip/hip_runtime.h>

// Problem dims (fixed by the reference)
#define B_ 512
#define H_ 1024
#define C_ 256
#define W_ 512

typedef __attribute__((ext_vector_type(2))) float v2f;
typedef __attribute__((ext_vector_type(8))) float v8f;

// ---------------------------------------------------------------------------
// Wave32 reductions
// ---------------------------------------------------------------------------
__device__ __forceinline__ float wave_max(float v) {
#pragma unroll
  for (int o = 16; o > 0; o >>= 1) v = fmaxf(v, __shfl_xor(v, o, 32));
  return v;
}
__device__ __forceinline__ float wave_sum(float v) {
#pragma unroll
  for (int o = 16; o > 0; o >>= 1) v += __shfl_xor(v, o, 32);
  return v;
}

// ---------------------------------------------------------------------------
// Kernel 1: logits_c = h_p @ psi_w.T   via V_WMMA_F32_16X16X4_F32
// One wave per 16x16 output tile. grid = (C/16, B/16), block = 32.
//
// A (16x4 f32, M x K): lanes 0-15 hold M=lane, VGPR0=K0,VGPR1=K1;
//                      lanes 16-31 hold M=lane-16, VGPR0=K2,VGPR1=K3.
// B (4x16 f32, K x N): symmetric, N on lanes, K split the same way.
// Both fragments are therefore a contiguous float2 at (row, k + 2*half).
// ---------------------------------------------------------------------------
__global__ __launch_bounds__(32) void pc_gemm_wmma(const float* __restrict__ hp,
                                                   const float* __restrict__ psi,
                                                   float* __restrict__ out) {
  const int lane = threadIdx.x & 31;
  const int half = lane >> 4;   // 0: K0..1, 1: K2..3
  const int l    = lane & 15;
  const int tileN = blockIdx.x; // over C
  const int tileM = blockIdx.y; // over B

  const float* arow = hp  + (size_t)(tileM * 16 + l) * H_ + 2 * half; // h_p row M
  const float* brow = psi + (size_t)(tileN * 16 + l) * H_ + 2 * half; // psi_w row N

  v8f acc = {};
#pragma unroll 4
  for (int k = 0; k < H_; k += 4) {
    v2f a = *(const v2f*)(arow + k);
    v2f b = *(const v2f*)(brow + k);
    // 8 args: (neg_a, A, neg_b, B, c_mod, C, reuse_a, reuse_b)
    acc = __builtin_amdgcn_wmma_f32_16x16x4_f32(
        false, a, false, b, (short)0, acc, false, false);
  }

  // C/D layout: VGPR r -> (M = r + 8*half, N = l)
  const int outM = tileM * 16 + half * 8;
  const int outN = tileN * 16 + l;
#pragma unroll
  for (int r = 0; r < 8; ++r)
    out[(size_t)(outM + r) * C_ + outN] = acc[r];
}

// ---------------------------------------------------------------------------
// Kernel 2: in-place row softmax over C=256 (p_c). grid = B, block = 256.
// ---------------------------------------------------------------------------
__global__ __launch_bounds__(256) void pc_softmax(float* __restrict__ pc) {
  __shared__ float red[8];
  const int b = blockIdx.x, tid = threadIdx.x;
  const int lane = tid & 31, wave = tid >> 5;

  float x = pc[(size_t)b * C_ + tid];

  float m = wave_max(x);
  if (lane == 0) red[wave] = m;
  __syncthreads();
  float bm = red[0];
#pragma unroll
  for (int i = 1; i < 8; ++i) bm = fmaxf(bm, red[i]);
  __syncthreads();

  float e = expf(x - bm);
  float s = wave_sum(e);
  if (lane == 0) red[wave] = s;
  __syncthreads();
  float bs = red[0];
#pragma unroll
  for (int i = 1; i < 8; ++i) bs += red[i];

  pc[(size_t)b * C_ + tid] = e / bs;
}

// ---------------------------------------------------------------------------
// Kernel 3: per-sample expert GEMV + masked softmax (bandwidth-dominant).
// grid = B, block = 256 (8 waves). h_p[b] staged in LDS; each wave dots 64
// rows of phi_w[cid] with coalesced float4 streaming loads + prefetch.
// ---------------------------------------------------------------------------
__global__ __launch_bounds__(256) void expert_pw(const float* __restrict__ hp,
                                                 const int*   __restrict__ tc,
                                                 const int*   __restrict__ cs,
                                                 const float* __restrict__ phi_w,
                                                 const float* __restrict__ phi_b,
                                                 float*       __restrict__ pw) {
  __shared__ __align__(16) float hsh[H_];
  __shared__ float lsh[W_];
  __shared__ float red[8];

  const int b = blockIdx.x, tid = threadIdx.x;
  const int lane = tid & 31, wave = tid >> 5;
  const int cid   = tc[b];
  const int csize = cs[cid];

  // Stage h_p[b] (4 KB) into LDS, fully coalesced float4.
  {
    const float4* src = (const float4*)(hp + (size_t)b * H_);
    ((float4*)hsh)[tid] = src[tid]; // 256 threads x 16 B = 4 KB
  }
  __syncthreads();

  const float* wbase = phi_w + (size_t)cid * W_ * H_;
  const float4* hv = (const float4*)hsh;

  // Each wave: rows [wave*64, wave*64+64). Per row: 32 lanes x 8 x float4.
  const int r0 = wave * 64, r1 = r0 + 64;
  for (int r = r0; r < r1; ++r) {
    const float4* wrow = (const float4*)(wbase + (size_t)r * H_);
    // Speculative prefetch 4 rows ahead (gfx1250 global_prefetch_b8);
    // lanes cover the 4 KB row at 128 B strides.
    __builtin_prefetch(wbase + (size_t)(r + 4) * H_ + lane * 32, 0, 0);

    float acc = 0.f;
#pragma unroll
    for (int i = 0; i < 8; ++i) {
      float4 w4 = wrow[lane + i * 32];
      float4 h4 = hv[lane + i * 32];
      acc = fmaf(w4.x, h4.x, acc);
      acc = fmaf(w4.y, h4.y, acc);
      acc = fmaf(w4.z, h4.z, acc);
      acc = fmaf(w4.w, h4.w, acc);
    }
    acc = wave_sum(acc);
    if (lane == 0) lsh[r] = acc + phi_b[(size_t)cid * W_ + r];
  }
  __syncthreads();

  // Masked filter (exact reference formula) + softmax over W=512, 2/thread.
  const int j0 = tid, j1 = tid + 256;
  float l0 = lsh[j0], l1 = lsh[j1];
  bool  v0 = j0 < csize, v1 = j1 < csize;
  float mp0 = v0 ? 1.f : -1.f,   mp1 = v1 ? 1.f : -1.f;
  float mn0 = v0 ? 1.f : -1e5f,  mn1 = v1 ? 1.f : -1e5f;
  float f0 = (l0 > 0.f ? l0 : l0 * mp0) * mn0;
  float f1 = (l1 > 0.f ? l1 : l1 * mp1) * mn1;

  float m = wave_max(fmaxf(f0, f1));
  if (lane == 0) red[wave] = m;
  __syncthreads();
  float bm = red[0];
#pragma unroll
  for (int i = 1; i < 8; ++i) bm = fmaxf(bm, red[i]);
  __syncthreads();

  float e0 = expf(f0 - bm), e1 = expf(f1 - bm);
  float s = wave_sum(e0 + e1);
  if (lane == 0) red[wave] = s;
  __syncthreads();
  float bs = red[0];
#pragma unroll
  for (int i = 1; i < 8; ++i) bs += red[i];

  pw[(size_t)b * W_ + j0] = e0 / bs;
  pw[(size_t)b * W_ + j1] = e1 / bs;
}

// ---------------------------------------------------------------------------
// Launch: inputs in setup_inputs order:
//   0 h_p [B,H] f32 | 1 target_cluster [B] i32 | 2 cluster_sizes [C] i32
//   3 psi_w [C,H] f32 | 4 phi_w [C,W,H] f32 | 5 phi_b [C,W] f32
// d_out = p_c [B*C] ++ p_w [B*W] (f32). Raw GEMM logits are written into the
// p_c region and softmaxed in place (stream-ordered), so no workspace needed.
// ---------------------------------------------------------------------------
extern "C" void kernel_launch(void* const* d_in, const int* in_sizes, int n_in,
                              void* d_out, int out_size, void* d_ws, size_t ws_size,
                              hipStream_t stream) {
  const float* hp    = (const float*)d_in[0];
  const int*   tcl   = (const int*)  d_in[1];
  const int*   csz   = (const int*)  d_in[2];
  const float* psi   = (const float*)d_in[3];
  const float* phi_w = (const float*)d_in[4];
  const float* phi_b = (const float*)d_in[5];

  float* pc = (float*)d_out;
  float* pw = pc + (size_t)B_ * C_;

  dim3 g1(C_ / 16, B_ / 16);                       // 16 x 32 tiles, 1 wave each
  pc_gemm_wmma<<<g1, 32, 0, stream>>>(hp, psi, pc);
  pc_softmax<<<B_, C_, 0, stream>>>(pc);
  expert_pw<<<B_, 256, 0, stream>>>(hp, tcl, csz, phi_w, phi_b, pw);
}